// PosMLPSelfAttention_8546984919447
// MI455X (gfx1250) — compile-verified
//
#include <hip/hip_runtime.h>

typedef __attribute__((ext_vector_type(2))) float v2f;
typedef __attribute__((ext_vector_type(8))) float v8f;

#define BATCH 2
#define SEQ   1024
#define DIM   1024
#define NH    16
#define HD    64
#define WGN   97    // 6*HID+1
#define WGP   112   // padded row stride for wg output

static __device__ __forceinline__ v8f wmma4(v2f a, v2f b, v8f c) {
  // D = A(16x4 f32) * B(4x16 f32) + C(16x16 f32)
  return __builtin_amdgcn_wmma_f32_16x16x4_f32(false, a, false, b, (short)0, c,
                                               false, false);
}

// ---------------------------------------------------------------------------
// Generic fp32 WMMA GEMM: C[M,ldc] = A[M,K] * B[K,N] (+ biasv[n]) (+ resid)
// One wave computes a 64(M) x 64(N) tile: 4 M-tiles x 4 N-tiles of 16x16.
// GUARD=false assumes N % 64 == 0 (no bounds checks in the hot loop).
// GUARD=true handles ragged N with a branchless clamp+mask (no exec waves).
// ---------------------------------------------------------------------------
template <bool GUARD>
__global__ __launch_bounds__(128) void gemm_wmma(
    const float* __restrict__ A, const float* __restrict__ Bm,
    const float* __restrict__ biasv, const float* __restrict__ resid,
    float* __restrict__ C, int M, int N, int K, int ldb, int ldc)
{
  const int lane = threadIdx.x;
  const int hi   = lane >> 4;      // half-wave select
  const int lq   = lane & 15;
  const int n0   = (blockIdx.x * blockDim.y + threadIdx.y) * 64;
  const int m0   = blockIdx.y * 64;
  if (n0 >= N) return;             // wave-uniform early exit (tail chunks)

  v8f acc[4][4];
#pragma unroll
  for (int i = 0; i < 4; ++i)
#pragma unroll
    for (int j = 0; j < 4; ++j)
      acc[i][j] = (v8f){0.f,0.f,0.f,0.f,0.f,0.f,0.f,0.f};

  const float* ap = A + (size_t)(m0 + lq) * K + 2 * hi;

#pragma unroll 2
  for (int k0 = 0; k0 < K; k0 += 4) {
    // A-frag layout: lanes0-15 hold K=k0,k0+1; lanes16-31 hold K=k0+2,k0+3
    v2f a[4];
#pragma unroll
    for (int i = 0; i < 4; ++i)
      a[i] = *(const v2f*)(ap + (size_t)i * 16 * K + k0);

    const float* brow = Bm + (size_t)(k0 + 2 * hi) * ldb;
#pragma unroll
    for (int j = 0; j < 4; ++j) {
      const int n = n0 + j * 16 + lq;
      v2f b;
      if (GUARD) {
        const int   ns  = (n < N) ? n : (N - 1);   // v_cndmask (no exec mask)
        const float msk = (n < N) ? 1.0f : 0.0f;
        b.x = brow[ns] * msk;
        b.y = brow[ldb + ns] * msk;
      } else {
        b.x = brow[n];         // B row k0+2*hi
        b.y = brow[ldb + n];   // B row k0+2*hi+1
      }
#pragma unroll
      for (int i = 0; i < 4; ++i)
        acc[i][j] = wmma4(a[i], b, acc[i][j]);
    }
  }

#pragma unroll
  for (int i2 = 0; i2 < 4; ++i2) {
#pragma unroll
    for (int j = 0; j < 4; ++j) {
      const int n = n0 + j * 16 + lq;
      if (!GUARD || n < N) {
        const float bv = biasv ? biasv[n] : 0.0f;
#pragma unroll
        for (int v = 0; v < 8; ++v) {
          const int row = m0 + i2 * 16 + 8 * hi + v;  // C-layout: vgpr + 8*hi
          float val = acc[i2][j][v] + bv;
          if (resid) val += resid[(size_t)row * N + n];
          C[(size_t)row * ldc + n] = val;
        }
      }
    }
  }
}

// ---------------------------------------------------------------------------
// Positional MLP bias: bias[b,i,j] = W2 . relu(feats(i,j) @ W1 + b1) + b2
// One block per (b,i); per-row params staged in LDS.
// ---------------------------------------------------------------------------
__global__ __launch_bounds__(256) void pos_bias_kernel(
    const float* __restrict__ pos, const float* __restrict__ wg,
    float* __restrict__ biasb)
{
  const int i = blockIdx.x;
  const int b = blockIdx.y;
  __shared__ float w[WGN];
  __shared__ float p1[4];
  for (int t = threadIdx.x; t < WGN; t += 256)
    w[t] = wg[(size_t)(b * SEQ + i) * WGP + t];
  if (threadIdx.x < 4)
    p1[threadIdx.x] = pos[(size_t)(b * SEQ + i) * 4 + threadIdx.x];
  __syncthreads();

  const float cx1 = p1[0], cy1 = p1[1], w1 = p1[2], h1 = p1[3];
  const float iw1 = 1.0f / w1, ih1 = 1.0f / h1;

  for (int j = threadIdx.x; j < SEQ; j += 256) {
    float4 p2 = *(const float4*)(pos + (size_t)(b * SEQ + j) * 4);
    float dx = (p2.x - cx1) * iw1;
    float dy = (p2.y - cy1) * ih1;
    float dw = __logf(p2.z * iw1);
    float dh = __logf(p2.w * ih1);
    float acc = w[96];
#pragma unroll
    for (int h = 0; h < 16; ++h) {
      float hid = dx * w[h] + dy * w[16 + h] + dw * w[32 + h] +
                  dh * w[48 + h] + w[64 + h];
      hid = fmaxf(hid, 0.0f);
      acc += hid * w[80 + h];
    }
    biasb[(size_t)(b * SEQ + i) * SEQ + j] = acc;
  }
}

// ---------------------------------------------------------------------------
// Flash attention, one wave per 16-query tile of one (b,h).
// Computes S^T = K*Q^T (keys x queries) so softmax stats are lane-local,
// then O^T = V^T * P^T with a shuffle-based C-layout -> B-frag remap.
// ---------------------------------------------------------------------------
__global__ __launch_bounds__(128) void attn_wmma(
    const float* __restrict__ qkv, const float* __restrict__ biasb,
    float* __restrict__ attnout)
{
  const int lane  = threadIdx.x;
  const int hi    = lane >> 4;
  const int lq    = lane & 15;
  const int qtile = blockIdx.x * blockDim.y + threadIdx.y;  // 0..63
  const int h     = blockIdx.y;
  const int b     = blockIdx.z;

  // Preload Q as B-fragments (Q^T): qa[t] serves k-step d=4t..4t+3
  const float* qbase =
      qkv + (size_t)(b * SEQ + qtile * 16 + lq) * (3 * DIM) + h * HD;
  v2f qa[16];
#pragma unroll
  for (int t = 0; t < 16; ++t)
    qa[t] = *(const v2f*)(qbase + 4 * t + 2 * hi);

  v8f oacc[4];
#pragma unroll
  for (int dt = 0; dt < 4; ++dt)
    oacc[dt] = (v8f){0.f,0.f,0.f,0.f,0.f,0.f,0.f,0.f};
  float m = -__builtin_inff();
  float l = 0.0f;

  for (int kt = 0; kt < SEQ / 16; ++kt) {
    // ---- S^T tile: 16 keys x 16 queries, reduce over d = 0..63 ----
    const float* kbase =
        qkv + (size_t)(b * SEQ + kt * 16 + lq) * (3 * DIM) + DIM + h * HD;
    v8f sacc = (v8f){0.f,0.f,0.f,0.f,0.f,0.f,0.f,0.f};
#pragma unroll
    for (int t = 0; t < 16; ++t) {
      v2f kf = *(const v2f*)(kbase + 4 * t + 2 * hi);  // A-frag of K
      sacc = wmma4(kf, qa[t], sacc);
    }

    // ---- logits^T = S^T * scale + bias; element (key v+8*hi, query lq) ----
    const float* bp =
        biasb + (size_t)(b * SEQ + qtile * 16 + lq) * SEQ + kt * 16 + 8 * hi;
    float pl[8];
    float lmax = -__builtin_inff();
#pragma unroll
    for (int v = 0; v < 8; ++v) {
      pl[v] = sacc[v] * 0.125f + bp[v];
      lmax = fmaxf(lmax, pl[v]);
    }
    lmax = fmaxf(lmax, __shfl_xor(lmax, 16, 32));  // combine halves (same q)
    float mnew = fmaxf(m, lmax);
    float rs = 0.0f;
#pragma unroll
    for (int v = 0; v < 8; ++v) {
      pl[v] = __expf(pl[v] - mnew);
      rs += pl[v];
    }
    rs += __shfl_xor(rs, 16, 32);
    float alpha = __expf(m - mnew);
    l = l * alpha + rs;
    m = mnew;
#pragma unroll
    for (int dt = 0; dt < 4; ++dt) oacc[dt] *= alpha;

    // ---- O^T += V^T(16d x 16key) * P^T(16key x 16q), 4 k-steps ----
#pragma unroll
    for (int t2 = 0; t2 < 4; ++t2) {
      const int e = (t2 & 1) * 4;  // pl[] base for this k-step
      v2f pf;                      // B-frag of P^T for keys 4*t2..4*t2+3
      if (t2 < 2) {
        float s2 = __shfl_xor(pl[e + 2], 16, 32);
        float s3 = __shfl_xor(pl[e + 3], 16, 32);
        pf.x = hi ? s2 : pl[e + 0];
        pf.y = hi ? s3 : pl[e + 1];
      } else {
        float s0 = __shfl_xor(pl[e + 0], 16, 32);
        float s1 = __shfl_xor(pl[e + 1], 16, 32);
        pf.x = hi ? pl[e + 2] : s0;
        pf.y = hi ? pl[e + 3] : s1;
      }
      const int kg = kt * 16 + 4 * t2 + 2 * hi;  // this lane's key pair
#pragma unroll
      for (int dt = 0; dt < 4; ++dt) {
        const float* vb = qkv + (size_t)(b * SEQ + kg) * (3 * DIM) +
                          2 * DIM + h * HD + dt * 16 + lq;
        v2f af;                    // A-frag of V^T: rows are d, K-dim is keys
        af.x = vb[0];
        af.y = vb[3 * DIM];
        oacc[dt] = wmma4(af, pf, oacc[dt]);
      }
    }
  }

  // ---- write O = (O^T)^T / l : element (d = dt*16+v+8*hi, q = lq) ----
  const float invl = 1.0f / l;
#pragma unroll
  for (int dt = 0; dt < 4; ++dt) {
    float* op = attnout + (size_t)(b * SEQ + qtile * 16 + lq) * DIM +
                h * HD + dt * 16 + 8 * hi;
#pragma unroll
    for (int v = 0; v < 8; ++v) op[v] = oacc[dt][v] * invl;
  }
}

// ---------------------------------------------------------------------------
// LayerNorm over last dim (1024) with gamma/beta; one block per row.
// ---------------------------------------------------------------------------
__global__ __launch_bounds__(256) void ln_kernel(
    const float* __restrict__ y, const float* __restrict__ g,
    const float* __restrict__ be, float* __restrict__ out)
{
  const int row = blockIdx.x;
  const float* yr = y + (size_t)row * DIM;
  float s = 0.f, q = 0.f;
  for (int c = threadIdx.x; c < DIM; c += 256) {
    float v = yr[c];
    s += v;
    q += v * v;
  }
#pragma unroll
  for (int off = 16; off; off >>= 1) {
    s += __shfl_xor(s, off, 32);
    q += __shfl_xor(q, off, 32);
  }
  __shared__ float ss[8], sq[8];
  const int wid = threadIdx.x >> 5;
  if ((threadIdx.x & 31) == 0) { ss[wid] = s; sq[wid] = q; }
  __syncthreads();
  float ts = 0.f, tq = 0.f;
#pragma unroll
  for (int i = 0; i < 8; ++i) { ts += ss[i]; tq += sq[i]; }
  const float mu  = ts * (1.0f / DIM);
  const float var = tq * (1.0f / DIM) - mu * mu;
  const float ri  = rsqrtf(var + 1e-5f);
  for (int c = threadIdx.x; c < DIM; c += 256)
    out[(size_t)row * DIM + c] = (yr[c] - mu) * ri * g[c] + be[c];
}

// ---------------------------------------------------------------------------
extern "C" void kernel_launch(void* const* d_in, const int* in_sizes, int n_in,
                              void* d_out, int out_size, void* d_ws,
                              size_t ws_size, hipStream_t stream)
{
  (void)in_sizes; (void)n_in; (void)out_size; (void)ws_size;
  const float* x     = (const float*)d_in[0];
  const float* pos   = (const float*)d_in[1];
  const float* qkvw  = (const float*)d_in[2];
  const float* outw  = (const float*)d_in[3];
  const float* wgw   = (const float*)d_in[4];
  const float* wgb   = (const float*)d_in[5];
  const float* gamma = (const float*)d_in[6];
  const float* beta  = (const float*)d_in[7];
  float* out = (float*)d_out;

  float* ws    = (float*)d_ws;
  float* qkvb  = ws;                                     // [2,1024,3072]
  float* wgo   = qkvb + (size_t)BATCH * SEQ * 3 * DIM;   // [2,1024,112]
  float* biasb = wgo  + (size_t)BATCH * SEQ * WGP;       // [2,1024,1024]
  float* attnb = biasb + (size_t)BATCH * SEQ * SEQ;      // [2,1024,1024]
  float* yb    = attnb + (size_t)BATCH * SEQ * DIM;      // [2048,1024]

  const dim3 blk(32, 4);
  const int M = BATCH * SEQ;  // 2048

  // qkv = x @ qkv_w                        (N=3072 -> 48 chunks / 4 waves)
  gemm_wmma<false><<<dim3(12, M / 64), blk, 0, stream>>>(
      x, qkvw, nullptr, nullptr, qkvb, M, 3 * DIM, DIM, 3 * DIM, 3 * DIM);
  // wg = x @ wg_w + wg_b                   (N=97, padded store stride 112)
  gemm_wmma<true><<<dim3(1, M / 64), blk, 0, stream>>>(
      x, wgw, wgb, nullptr, wgo, M, WGN, DIM, WGN, WGP);
  // positional MLP bias
  pos_bias_kernel<<<dim3(SEQ, BATCH), 256, 0, stream>>>(pos, wgo, biasb);
  // attention (flash, transposed tiles)
  attn_wmma<<<dim3(16, NH, BATCH), blk, 0, stream>>>(qkvb, biasb, attnb);
  // y = attn @ out_w + x
  gemm_wmma<false><<<dim3(4, M / 64), blk, 0, stream>>>(
      attnb, outw, nullptr, x, yb, M, DIM, DIM, DIM, DIM);
  // layernorm -> output
  ln_kernel<<<M, 256, 0, stream>>>(yb, gamma, beta, out);
}